// LSTMLayer_68066641707649
// MI455X (gfx1250) — compile-verified
//
#include <hip/hip_runtime.h>

typedef float v2f __attribute__((ext_vector_type(2)));
typedef float v8f __attribute__((ext_vector_type(8)));

#define T_DIM 512
#define B_DIM 128
#define I_DIM 512
#define H_DIM 512
#define G4H   (4 * H_DIM)

__device__ __forceinline__ float sigmoidf_(float x) {
    return 1.0f / (1.0f + __expf(-x));
}

__device__ __forceinline__ v2f ld2(const float* p) {
    return *(const v2f*)p;
}

// ---------------------------------------------------------------------------
// Phase 1 (parallel over T): pre[t*B+m, n] = X[t*B+m, :] . W_ih[n, :]
// One wave computes a 16x64 tile (4 WMMA n-tiles) to reuse the A operand 4x.
// WMMA f32 16x16x4: A 16x4 (v2f/lane), B 4x16 (v2f/lane), C/D 16x16 (v8f/lane)
//   A: lane = {lm=M, lh}, VGPR0/1 = K = 2*lh + {0,1}          -> float2 along K
//   B: lane = {lm=N, lh}, VGPR0/1 = K = 2*lh + {0,1}          -> float2 along K (Wt col = W row)
//   D: lane = {lm=N, lh}, VGPR r   = M = r + 8*lh
// ---------------------------------------------------------------------------
__global__ void __launch_bounds__(256) xw_gemm_kernel(
    const float* __restrict__ X,    // [T*B, I]
    const float* __restrict__ Wih,  // [4H, I]
    float* __restrict__ pre)        // [T*B, 4H]
{
    const int lane = threadIdx.x & 31;
    const int wid  = blockIdx.x * (blockDim.x >> 5) + (threadIdx.x >> 5);
    const int mt = wid >> 5;            // 0 .. T*B/16 - 1
    const int ng = wid & 31;            // 0 .. 31 (groups of 64 cols)
    const int m0 = mt << 4;
    const int n0 = ng << 6;
    const int lm = lane & 15;
    const int lh = lane >> 4;

    const float* ar  = X   + (size_t)(m0 + lm) * I_DIM + 2 * lh;
    const float* br0 = Wih + (size_t)(n0 +  0 + lm) * I_DIM + 2 * lh;
    const float* br1 = Wih + (size_t)(n0 + 16 + lm) * I_DIM + 2 * lh;
    const float* br2 = Wih + (size_t)(n0 + 32 + lm) * I_DIM + 2 * lh;
    const float* br3 = Wih + (size_t)(n0 + 48 + lm) * I_DIM + 2 * lh;

    v8f acc0 = {}, acc1 = {}, acc2 = {}, acc3 = {};
#pragma unroll 4
    for (int k = 0; k < I_DIM; k += 4) {
        v2f a  = ld2(ar  + k);
        v2f b0 = ld2(br0 + k);
        v2f b1 = ld2(br1 + k);
        v2f b2 = ld2(br2 + k);
        v2f b3 = ld2(br3 + k);
        acc0 = __builtin_amdgcn_wmma_f32_16x16x4_f32(false, a, false, b0, (short)0, acc0, false, false);
        acc1 = __builtin_amdgcn_wmma_f32_16x16x4_f32(false, a, false, b1, (short)0, acc1, false, false);
        acc2 = __builtin_amdgcn_wmma_f32_16x16x4_f32(false, a, false, b2, (short)0, acc2, false, false);
        acc3 = __builtin_amdgcn_wmma_f32_16x16x4_f32(false, a, false, b3, (short)0, acc3, false, false);
    }

#pragma unroll
    for (int r = 0; r < 8; ++r) {
        size_t row = (size_t)(m0 + r + 8 * lh) * G4H + n0 + lm;
        pre[row +  0] = acc0[r];
        pre[row + 16] = acc1[r];
        pre[row + 32] = acc2[r];
        pre[row + 48] = acc3[r];
    }
}

// ---------------------------------------------------------------------------
// One recurrent step. Each wave owns one 16(batch) x 16(hidden) tile of h/c
// and computes all 4 gate tiles for it (A operand h-tile reused 4x).
// USE_PRE: x@W_ih^T already in `pre_t`; otherwise fused here.
// FIRST:   h_prev == 0 (skip W_hh GEMM) and c_prev == 0.
// ---------------------------------------------------------------------------
template <bool USE_PRE, bool FIRST>
__global__ void __launch_bounds__(128) lstm_step_kernel(
    const float* __restrict__ xt,     // [B, I]   (only if !USE_PRE)
    const float* __restrict__ hprev,  // [B, H]   (only if !FIRST)
    const float* __restrict__ Wih,    // [4H, I]  (only if !USE_PRE)
    const float* __restrict__ Whh,    // [4H, H]
    const float* __restrict__ bih,    // [4H]
    const float* __restrict__ bhh,    // [4H]
    const float* __restrict__ pre_t,  // [B, 4H]  (only if USE_PRE)
    float* __restrict__ cstate,       // [B, H]
    float* __restrict__ hout)         // [B, H]
{
    const int lane = threadIdx.x & 31;
    const int wid  = blockIdx.x * 4 + (threadIdx.x >> 5);  // 0..255
    const int bt = wid >> 5;            // 0..7
    const int nt = wid & 31;            // 0..31
    const int b0 = bt << 4;
    const int n0 = nt << 4;
    const int lm = lane & 15;
    const int lh = lane >> 4;
    const int col = n0 + lm;

    v8f zero = {};
    v8f acc[4] = {zero, zero, zero, zero};

    if (!FIRST) {  // h @ W_hh^T  (K = H, W_hh L2-resident)
        const float* ar  = hprev + (size_t)(b0 + lm) * H_DIM + 2 * lh;
        const float* br0 = Whh + (size_t)(0 * H_DIM + col) * H_DIM + 2 * lh;
        const float* br1 = Whh + (size_t)(1 * H_DIM + col) * H_DIM + 2 * lh;
        const float* br2 = Whh + (size_t)(2 * H_DIM + col) * H_DIM + 2 * lh;
        const float* br3 = Whh + (size_t)(3 * H_DIM + col) * H_DIM + 2 * lh;
#pragma unroll 4
        for (int k = 0; k < H_DIM; k += 4) {
            v2f a  = ld2(ar  + k);
            v2f b0v = ld2(br0 + k);
            v2f b1v = ld2(br1 + k);
            v2f b2v = ld2(br2 + k);
            v2f b3v = ld2(br3 + k);
            acc[0] = __builtin_amdgcn_wmma_f32_16x16x4_f32(false, a, false, b0v, (short)0, acc[0], false, false);
            acc[1] = __builtin_amdgcn_wmma_f32_16x16x4_f32(false, a, false, b1v, (short)0, acc[1], false, false);
            acc[2] = __builtin_amdgcn_wmma_f32_16x16x4_f32(false, a, false, b2v, (short)0, acc[2], false, false);
            acc[3] = __builtin_amdgcn_wmma_f32_16x16x4_f32(false, a, false, b3v, (short)0, acc[3], false, false);
        }
    }

    if (!USE_PRE) {  // fused x_t @ W_ih^T  (K = I)
        const float* ar  = xt + (size_t)(b0 + lm) * I_DIM + 2 * lh;
        const float* br0 = Wih + (size_t)(0 * H_DIM + col) * I_DIM + 2 * lh;
        const float* br1 = Wih + (size_t)(1 * H_DIM + col) * I_DIM + 2 * lh;
        const float* br2 = Wih + (size_t)(2 * H_DIM + col) * I_DIM + 2 * lh;
        const float* br3 = Wih + (size_t)(3 * H_DIM + col) * I_DIM + 2 * lh;
#pragma unroll 4
        for (int k = 0; k < I_DIM; k += 4) {
            v2f a  = ld2(ar  + k);
            v2f b0v = ld2(br0 + k);
            v2f b1v = ld2(br1 + k);
            v2f b2v = ld2(br2 + k);
            v2f b3v = ld2(br3 + k);
            acc[0] = __builtin_amdgcn_wmma_f32_16x16x4_f32(false, a, false, b0v, (short)0, acc[0], false, false);
            acc[1] = __builtin_amdgcn_wmma_f32_16x16x4_f32(false, a, false, b1v, (short)0, acc[1], false, false);
            acc[2] = __builtin_amdgcn_wmma_f32_16x16x4_f32(false, a, false, b2v, (short)0, acc[2], false, false);
            acc[3] = __builtin_amdgcn_wmma_f32_16x16x4_f32(false, a, false, b3v, (short)0, acc[3], false, false);
        }
    } else {  // add precomputed x@W_ih^T contribution
#pragma unroll
        for (int r = 0; r < 8; ++r) {
            size_t row = (size_t)(b0 + r + 8 * lh) * G4H + col;
            acc[0][r] += pre_t[row + 0 * H_DIM];
            acc[1][r] += pre_t[row + 1 * H_DIM];
            acc[2][r] += pre_t[row + 2 * H_DIM];
            acc[3][r] += pre_t[row + 3 * H_DIM];
        }
    }

    float bs0 = bih[0 * H_DIM + col] + bhh[0 * H_DIM + col];
    float bs1 = bih[1 * H_DIM + col] + bhh[1 * H_DIM + col];
    float bs2 = bih[2 * H_DIM + col] + bhh[2 * H_DIM + col];
    float bs3 = bih[3 * H_DIM + col] + bhh[3 * H_DIM + col];

#pragma unroll
    for (int r = 0; r < 8; ++r) {
        int m = b0 + r + 8 * lh;
        size_t idx = (size_t)m * H_DIM + col;
        float ig = sigmoidf_(acc[0][r] + bs0);
        float fg = sigmoidf_(acc[1][r] + bs1);
        float gg = tanhf(acc[2][r] + bs2);
        float og = sigmoidf_(acc[3][r] + bs3);
        float cprev = FIRST ? 0.0f : cstate[idx];
        float cn = fg * cprev + ig * gg;
        cstate[idx] = cn;
        hout[idx] = og * tanhf(cn);
    }
}

extern "C" void kernel_launch(void* const* d_in, const int* in_sizes, int n_in,
                              void* d_out, int out_size, void* d_ws, size_t ws_size,
                              hipStream_t stream) {
    (void)in_sizes; (void)n_in; (void)out_size;

    const float* X   = (const float*)d_in[0];  // [T, B, I]
    // d_in[1] = time, unused by the base cell
    const float* Wih = (const float*)d_in[2];  // [4H, I]
    const float* Whh = (const float*)d_in[3];  // [4H, H]
    const float* bih = (const float*)d_in[4];  // [4H]
    const float* bhh = (const float*)d_in[5];  // [4H]

    float* out = (float*)d_out;                               // outputs [T, B, H]
    float* hT  = out + (size_t)T_DIM * B_DIM * H_DIM;         // then hT [B, H]
    float* cT  = hT + (size_t)B_DIM * H_DIM;                  // then cT [B, H]

    const size_t cBytes   = (size_t)B_DIM * H_DIM * sizeof(float);
    const size_t preBytes = (size_t)T_DIM * B_DIM * G4H * sizeof(float);
    float* c   = (float*)d_ws;
    float* pre = (float*)((char*)d_ws + cBytes);
    const bool usePre = (ws_size >= cBytes + preBytes);

    if (usePre) {
        // M = T*B = 65536 -> 4096 m-tiles; N = 2048 -> 32 groups of 64 cols.
        // 131072 waves, 8 waves/block -> 16384 blocks.
        xw_gemm_kernel<<<16384, 256, 0, stream>>>(X, Wih, pre);
    }

    for (int t = 0; t < T_DIM; ++t) {
        const float* xt    = X + (size_t)t * B_DIM * I_DIM;
        const float* hprev = (t == 0) ? nullptr : out + (size_t)(t - 1) * B_DIM * H_DIM;
        const float* pret  = pre + (size_t)t * B_DIM * G4H;
        float* hout        = out + (size_t)t * B_DIM * H_DIM;
        if (usePre) {
            if (t == 0)
                lstm_step_kernel<true, true><<<64, 128, 0, stream>>>(
                    nullptr, nullptr, nullptr, Whh, bih, bhh, pret, c, hout);
            else
                lstm_step_kernel<true, false><<<64, 128, 0, stream>>>(
                    nullptr, hprev, nullptr, Whh, bih, bhh, pret, c, hout);
        } else {
            if (t == 0)
                lstm_step_kernel<false, true><<<64, 128, 0, stream>>>(
                    xt, nullptr, Wih, Whh, bih, bhh, nullptr, c, hout);
            else
                lstm_step_kernel<false, false><<<64, 128, 0, stream>>>(
                    xt, hprev, Wih, Whh, bih, bhh, nullptr, c, hout);
        }
    }

    // (hT, cT): hT == outputs[T-1]; cT lives in scratch.
    hipMemcpyAsync(hT, out + (size_t)(T_DIM - 1) * B_DIM * H_DIM, cBytes,
                   hipMemcpyDeviceToDevice, stream);
    hipMemcpyAsync(cT, c, cBytes, hipMemcpyDeviceToDevice, stream);
}